// Temporal_FC1_GAP_action_89833535963820
// MI455X (gfx1250) — compile-verified
//
#include <hip/hip_runtime.h>

typedef float f32x4 __attribute__((ext_vector_type(4)));
typedef float v2f   __attribute__((ext_vector_type(2)));
typedef float v8f   __attribute__((ext_vector_type(8)));

#define B_   256
#define T_   300
#define C_   1152
#define O_   285
#define SEG  4
#define TSEG (T_ / SEG)   // 75

// ---------------------------------------------------------------------------
// Phase 1: streaming partial reduction over T (bandwidth bound, ~354 MB read).
// grid (B, SEG), 288 threads/block; each thread owns 4 consecutive channels
// and sums 75 timesteps with non-temporal b128 loads (stream-once data).
// ---------------------------------------------------------------------------
__global__ void reduce_partial_kernel(const float* __restrict__ x,
                                      float* __restrict__ part) {
  const int b   = blockIdx.x;
  const int seg = blockIdx.y;
  const int c   = threadIdx.x * 4;                    // 0..1148
  const float* base = x + ((size_t)b * T_ + (size_t)seg * TSEG) * C_ + c;

  f32x4 acc = {0.f, 0.f, 0.f, 0.f};
#pragma unroll 5
  for (int t = 0; t < TSEG; ++t) {
    f32x4 v = __builtin_nontemporal_load((const f32x4*)(base + (size_t)t * C_));
    acc += v;
  }
  *(f32x4*)(part + ((size_t)b * SEG + seg) * C_ + c) = acc;
}

// ---------------------------------------------------------------------------
// Phase 1b: combine the 4 T-segments, apply /64 and -2T  ->  z[b,c]
// ---------------------------------------------------------------------------
__global__ void combine_kernel(const float* __restrict__ part,
                               float* __restrict__ z) {
  const int C4 = C_ / 4;
  int e4 = blockIdx.x * blockDim.x + threadIdx.x;     // float4 index into z
  if (e4 >= B_ * C4) return;
  int b  = e4 / C4;
  int c  = (e4 - b * C4) * 4;

  f32x4 s = {0.f, 0.f, 0.f, 0.f};
#pragma unroll
  for (int seg = 0; seg < SEG; ++seg)
    s += *(const f32x4*)(part + ((size_t)b * SEG + seg) * C_ + c);

  f32x4 zz = s * (1.0f / 64.0f) - 2.0f * (float)T_;
  *(f32x4*)(z + (size_t)b * C_ + c) = zz;
}

// ---------------------------------------------------------------------------
// Phase 2: out[b,o] = (z[b,:]·W[o,:] + bias[o]*T) / length[b]
// One wave per 16x16 tile, f32 WMMA (16x16x4), K loop over C=1152 (288 MACs).
// Edge columns (O=285 < 288) read an address-clamped valid W row; their
// results are finite garbage that is simply never stored — keeps the inner
// loop at 2 loads + 1 WMMA with EXEC all-ones throughout.
// ---------------------------------------------------------------------------
__global__ void gemm_wmma_kernel(const float* __restrict__ z,
                                 const float* __restrict__ W,
                                 const float* __restrict__ bias,
                                 const int*   __restrict__ length,
                                 float* __restrict__ out) {
  const int lane  = threadIdx.x;        // 0..31, wave32
  const int b0    = blockIdx.x * 16;    // batch-tile origin (16 exact tiles)
  const int o0    = blockIdx.y * 16;    // output-channel tile origin (18 tiles)
  const int mn    = lane & 15;          // M index for A, N index for B/D
  const int khalf = lane >> 4;          // 0: K={0,1}, 1: K={2,3}
  const int koff  = khalf * 2;

  const int  oc     = o0 + mn;
  const int  ocl    = oc < O_ ? oc : (O_ - 1);        // clamp addr in-range
  const bool ovalid = (oc < O_);

  const float* arow = z + (size_t)(b0 + mn) * C_;     // A: row b0+mn of z
  const float* brow = W + (size_t)ocl * C_;           // B: row oc of W

  v8f acc = {0.f, 0.f, 0.f, 0.f, 0.f, 0.f, 0.f, 0.f};

  for (int kk = 0; kk < C_; kk += 4) {
    v2f a = *(const v2f*)(arow + kk + koff);
    v2f w = *(const v2f*)(brow + kk + koff);
    // D = A(16x4,f32) * B(4x16,f32) + C(16x16,f32)
    acc = __builtin_amdgcn_wmma_f32_16x16x4_f32(
        /*neg_a=*/false, a, /*neg_b=*/false, w,
        /*c_mod=*/(short)0, acc, /*reuse_a=*/false, /*reuse_b=*/false);
  }

  const float bt = (ovalid ? bias[oc] : 0.f) * (float)T_;
#pragma unroll
  for (int j = 0; j < 8; ++j) {
    const int gb   = b0 + j + 8 * khalf;              // D: VGPR j -> row j(+8)
    const float il = 1.0f / (float)length[gb];
    if (ovalid) out[(size_t)gb * O_ + oc] = (acc[j] + bt) * il;
  }
}

// ---------------------------------------------------------------------------
extern "C" void kernel_launch(void* const* d_in, const int* in_sizes, int n_in,
                              void* d_out, int out_size, void* d_ws, size_t ws_size,
                              hipStream_t stream) {
  const float* x      = (const float*)d_in[0];  // [B,T,C] f32
  const int*   length = (const int*)  d_in[1];  // [B] i32
  const float* W      = (const float*)d_in[2];  // [O,C] f32
  const float* bias   = (const float*)d_in[3];  // [O] f32
  float* out = (float*)d_out;                   // [B,O] f32

  float* part = (float*)d_ws;                       // SEG*B*C floats (4.7 MB)
  float* z    = part + (size_t)SEG * B_ * C_;       // B*C floats (1.2 MB)

  reduce_partial_kernel<<<dim3(B_, SEG), dim3(C_ / 4), 0, stream>>>(x, part);

  const int n4 = B_ * (C_ / 4);
  combine_kernel<<<dim3((n4 + 255) / 256), dim3(256), 0, stream>>>(part, z);

  gemm_wmma_kernel<<<dim3(B_ / 16, (O_ + 15) / 16), dim3(32), 0, stream>>>(
      z, W, bias, length, out);
}